// SCPDTLoss_24859270709589
// MI455X (gfx1250) — compile-verified
//
#include <hip/hip_runtime.h>
#include <math.h>

typedef __attribute__((ext_vector_type(2))) float v2f;
typedef __attribute__((ext_vector_type(8))) float v8f;

#define DD   12288           // d = C*P*P
#define KK   16
#define NPAT 512             // B*N
#define GC   22583.961392f   // d * ln(2*pi) = 12288 * 1.8378770664

// workspace layout (floats)
#define OFF_M    0                      // 512 * 256  VtDiV tiles
#define OFF_R    (NPAT*256)             // 512 * 16   Vt D^-1 r
#define OFF_V2   (OFF_R   + NPAT*16)    // 512        sum V^2
#define OFF_RDR  (OFF_V2  + NPAT)       // 512        r^T D^-1 r
#define OFF_LD   (OFF_RDR + NPAT)       // 512        sum log sigma
#define OFF_SS   (OFF_LD  + NPAT)       // 512        sum sigma
#define OFF_RL   (OFF_SS  + NPAT)       // 512        sum relu(0.3-sig)^2
#define OFF_MIN  (OFF_RL  + NPAT)       // 512        min sigma
#define OFF_BND  (OFF_MIN + NPAT)       // 64         boundary partials
#define NBND 64

__device__ __forceinline__ float block_sum256(float v, float* red) {
    int t = threadIdx.x;
    red[t] = v; __syncthreads();
    for (int o = 128; o; o >>= 1) { if (t < o) red[t] += red[t + o]; __syncthreads(); }
    float r = red[0]; __syncthreads();
    return r;
}
__device__ __forceinline__ float block_min256(float v, float* red) {
    int t = threadIdx.x;
    red[t] = v; __syncthreads();
    for (int o = 128; o; o >>= 1) { if (t < o) red[t] = fminf(red[t], red[t + o]); __syncthreads(); }
    float r = red[0]; __syncthreads();
    return r;
}

// x0 flat index for element e of patch (b,n):  (B,C,H,W) with P=64, gh=gw=8
__device__ __forceinline__ size_t x0_index(int b, int gy, int gx, int e) {
    int c  = e >> 12;
    int py = (e >> 6) & 63;
    int px = e & 63;
    return (((size_t)(b * 3 + c) * 512) + (size_t)(gy * 64 + py)) * 512 + (size_t)(gx * 64 + px);
}

// ---------------- Kernel 1: WMMA pass over V ----------------
// One patch per block, 4 waves each covering d/4 = 3072 rows.
// Per 4 d-rows: 1x V_WMMA_F32_16X16X4_F32 accumulating VtDiV (A = D^-1-scaled V^T slab, B = V slab).
__global__ __launch_bounds__(128) void k1_wmma(const float* __restrict__ x0,
                                               const float* __restrict__ mu,
                                               const float* __restrict__ sigma,
                                               const float* __restrict__ V,
                                               float* __restrict__ ws) {
    const int p    = blockIdx.x;
    const int b    = p >> 6, n = p & 63;
    const int gy   = n >> 3, gx = n & 7;
    const int wave = threadIdx.x >> 5;
    const int lane = threadIdx.x & 31;
    const int h    = lane >> 4;       // lane half selects k-pair / M-offset
    const int col  = lane & 15;       // m (for A) / n (for B)

    const float* Vp = V     + (size_t)p * (DD * KK);
    const float* sp = sigma + (size_t)p * DD;
    const float* mp = mu    + (size_t)p * DD;

    v8f  acc  = {};
    float vtdr = 0.f, v2 = 0.f;

    const int dstart = wave * (DD / 4);
    const int dend   = dstart + (DD / 4);

    #pragma unroll 2
    for (int d0 = dstart; d0 < dend; d0 += 4) {
        const int r0 = d0 + 2 * h;                       // this lane's first d-row
        float vb0 = Vp[(size_t)r0 * KK + col];           // V[r0][col]
        float vb1 = Vp[(size_t)(r0 + 1) * KK + col];     // V[r0+1][col]
        v2f sg = *(const v2f*)(sp + r0);                 // r0 even -> 8B aligned
        v2f mm = *(const v2f*)(mp + r0);
        v2f xx = *(const v2f*)(x0 + x0_index(b, gy, gx, r0)); // px even -> contiguous & aligned

        float di0 = __builtin_amdgcn_rcpf(sg.x);
        float di1 = __builtin_amdgcn_rcpf(sg.y);
        float a0 = vb0 * di0, a1 = vb1 * di1;

        v2f A  = {a0, a1};       // A[m=col][k] = Dinv * V (16x4 layout)
        v2f Bm = {vb0, vb1};     // B[k][n=col] = V        (4x16 layout)
        acc = __builtin_amdgcn_wmma_f32_16x16x4_f32(false, A, false, Bm,
                                                    (short)0, acc, false, false);

        vtdr = fmaf(a0, xx.x - mm.x, fmaf(a1, xx.y - mm.y, vtdr));
        v2   = fmaf(vb0, vb0, fmaf(vb1, vb1, v2));
    }

    // vtdr[m=col] = lane + lane^16 partner (covers all d mod 4 classes)
    vtdr += __shfl_xor(vtdr, 16, 32);
    for (int o = 16; o; o >>= 1) v2 += __shfl_xor(v2, o, 32);

    __shared__ float sM[4][256];
    __shared__ float sR[4][16];
    __shared__ float sV2[4];
    #pragma unroll
    for (int g = 0; g < 8; ++g)                 // C layout: M = g + 8*h, N = col
        sM[wave][(g + 8 * h) * 16 + col] = acc[g];
    if (h == 0)     sR[wave][col] = vtdr;
    if (lane == 0)  sV2[wave] = v2;
    __syncthreads();

    for (int i = threadIdx.x; i < 256; i += 128)
        ws[OFF_M + p * 256 + i] = (sM[0][i] + sM[1][i]) + (sM[2][i] + sM[3][i]);
    if (threadIdx.x < 16)
        ws[OFF_R + p * 16 + threadIdx.x] = (sR[0][threadIdx.x] + sR[1][threadIdx.x]) +
                                           (sR[2][threadIdx.x] + sR[3][threadIdx.x]);
    if (threadIdx.x == 0)
        ws[OFF_V2 + p] = (sV2[0] + sV2[1]) + (sV2[2] + sV2[3]);
}

// ---------------- Kernel 2: elementwise stats per patch ----------------
__global__ __launch_bounds__(256) void k2_scalar(const float* __restrict__ x0,
                                                 const float* __restrict__ mu,
                                                 const float* __restrict__ sigma,
                                                 float* __restrict__ ws) {
    __shared__ float red[256];
    const int p = blockIdx.x;
    const int b = p >> 6, n = p & 63, gy = n >> 3, gx = n & 7;
    const float* sp = sigma + (size_t)p * DD;
    const float* mp = mu    + (size_t)p * DD;

    float rdr = 0.f, ld = 0.f, ss = 0.f, rl = 0.f, mn = 3.4e38f;
    for (int e = threadIdx.x; e < DD; e += 256) {
        float sg = sp[e];
        float r  = x0[x0_index(b, gy, gx, e)] - mp[e];
        rdr = fmaf(r * r, __builtin_amdgcn_rcpf(sg), rdr);
        ld += __logf(sg);
        ss += sg;
        mn  = fminf(mn, sg);
        float t = 0.3f - sg;
        if (t > 0.f) rl = fmaf(t, t, rl);
    }
    float R;
    R = block_sum256(rdr, red); if (threadIdx.x == 0) ws[OFF_RDR + p] = R;
    R = block_sum256(ld,  red); if (threadIdx.x == 0) ws[OFF_LD  + p] = R;
    R = block_sum256(ss,  red); if (threadIdx.x == 0) ws[OFF_SS  + p] = R;
    R = block_sum256(rl,  red); if (threadIdx.x == 0) ws[OFF_RL  + p] = R;
    R = block_min256(mn,  red); if (threadIdx.x == 0) ws[OFF_MIN + p] = R;
}

// ---------------- Kernel 3: boundary loss on unpatchified mu ----------------
__device__ __forceinline__ float img_mu(const float* mu, int b, int c, int y, int x) {
    int n  = (y >> 6) * 8 + (x >> 6);
    int dd = c * 4096 + (y & 63) * 64 + (x & 63);
    return mu[((size_t)(b * 64 + n)) * DD + dd];
}

__global__ __launch_bounds__(256) void k3_boundary(const float* __restrict__ mu,
                                                   float* __restrict__ ws) {
    __shared__ float red[256];
    const int half  = 7 * 8 * 3 * 512;   // 86016 per direction
    const int total = 2 * half;
    float s = 0.f;
    for (int i = blockIdx.x * 256 + threadIdx.x; i < total; i += 256 * NBND) {
        int ii = i;
        bool isdy = ii < half;
        if (!isdy) ii -= half;
        int li   = ii / 12288;      // boundary 0..6
        int rem  = ii % 12288;
        int b    = rem / 1536;
        int rem2 = rem % 1536;
        int c    = rem2 / 512;
        int t    = rem2 % 512;
        float v;
        if (isdy) { int y = (li + 1) * 64; v = img_mu(mu, b, c, y, t) - img_mu(mu, b, c, y - 1, t); }
        else      { int x = (li + 1) * 64; v = img_mu(mu, b, c, t, x) - img_mu(mu, b, c, t, x - 1); }
        s = fmaf(v, v, s);
    }
    float R = block_sum256(s, red);
    if (threadIdx.x == 0) ws[OFF_BND + blockIdx.x] = R;
}

// ---------------- Kernel 4: Cholesky + final reduction ----------------
__global__ __launch_bounds__(256) void k4_final(float* __restrict__ ws,
                                                float* __restrict__ out) {
    __shared__ float red[256];
    const int t = threadIdx.x;

    float v2t  = block_sum256(ws[OFF_V2  + t] + ws[OFF_V2  + t + 256], red);
    float sst  = block_sum256(ws[OFF_SS  + t] + ws[OFF_SS  + t + 256], red);
    float rlt  = block_sum256(ws[OFF_RL  + t] + ws[OFF_RL  + t + 256], red);
    float mnt  = block_min256(fminf(ws[OFF_MIN + t], ws[OFF_MIN + t + 256]), red);
    float bndt = block_sum256(t < NBND ? ws[OFF_BND + t] : 0.f, red);

    float nll = 0.f;
    for (int p = t; p < NPAT; p += 256) {
        const float* Mp = ws + OFF_M + p * 256;
        const float* Rp = ws + OFF_R + p * 16;
        float L[16][16];
        for (int i = 0; i < 16; ++i) {
            for (int j = 0; j <= i; ++j) {
                float s = Mp[i * 16 + j] + (i == j ? (1.0f + 1e-6f) : 0.f);
                for (int q = 0; q < j; ++q) s -= L[i][q] * L[j][q];
                if (i == j) L[i][i] = sqrtf(s);
                else        L[i][j] = s / L[j][j];
            }
        }
        float y[16], ldM = 0.f, corr = 0.f;
        for (int i = 0; i < 16; ++i) {
            float s = Rp[i];
            for (int q = 0; q < i; ++q) s -= L[i][q] * y[q];
            y[i] = s / L[i][i];
            corr = fmaf(y[i], y[i], corr);
            ldM += 2.f * __logf(L[i][i]);
        }
        float quad = ws[OFF_RDR + p] - corr;
        nll += 0.5f * (quad + ws[OFF_LD + p] + ldM + GC) * (1.0f / (float)DD);
    }
    float nllt = block_sum256(nll, red);

    if (t == 0) {
        float recon  = nllt * (1.0f / 512.0f);
        float bnd    = bndt * (1.0f / (12288.0f * 14.0f));
        float rank   = v2t  * (1.0f / 100663296.0f);   // B*N*d*K
        float sigpen = rlt  * (1.0f / 6291456.0f);     // B*N*d
        float smean  = sst  * (1.0f / 6291456.0f);
        out[0] = recon + 0.1f * bnd + 0.01f * rank + 0.05f * sigpen;
        out[1] = recon;
        out[2] = bnd;
        out[3] = rank;
        out[4] = sigpen;
        out[5] = smean;
        out[6] = mnt;
    }
}

extern "C" void kernel_launch(void* const* d_in, const int* in_sizes, int n_in,
                              void* d_out, int out_size, void* d_ws, size_t ws_size,
                              hipStream_t stream) {
    const float* x0    = (const float*)d_in[0];
    const float* mu    = (const float*)d_in[1];
    const float* sigma = (const float*)d_in[2];
    const float* V     = (const float*)d_in[3];
    float* ws  = (float*)d_ws;
    float* out = (float*)d_out;

    k1_wmma    <<<NPAT, 128, 0, stream>>>(x0, mu, sigma, V, ws);
    k2_scalar  <<<NPAT, 256, 0, stream>>>(x0, mu, sigma, ws);
    k3_boundary<<<NBND, 256, 0, stream>>>(mu, ws);
    k4_final   <<<1,    256, 0, stream>>>(ws, out);
}